// ElmanNetwork_64141041599041
// MI455X (gfx1250) — compile-verified
//
#include <hip/hip_runtime.h>
#include <math.h>

typedef __attribute__((ext_vector_type(2))) float v2f;
typedef __attribute__((ext_vector_type(8))) float v8f;

#define VOCABN 50257
#define EMBN   300
#define HSZ    300     // hidden size (== EMB)
#define HIDN   512
#define NCLSN  4
#define BN     128
#define TN     512
#define ZSTR   304     // padded row stride (multiple of 16) for Z and H
#define NT1    19      // N tiles covering 304 (hidden dim, padded)
#define NT2    32      // N tiles covering 512 (HID)

static __device__ __forceinline__ v8f wmma4(v2f a, v2f b, v8f c) {
  // D = A(16x4, f32) * B(4x16, f32) + C(16x16, f32)
  return __builtin_amdgcn_wmma_f32_16x16x4_f32(
      false, a, false, b, (short)0, c, false, false);
}

// ---------------------------------------------------------------------------
// Zero h0 (ping state) and the relu-max accumulator.
__global__ void elman_init_kernel(float* __restrict__ h0,
                                  float* __restrict__ acc) {
  int i = blockIdx.x * blockDim.x + threadIdx.x;
  if (i < BN * ZSTR) h0[i] = 0.0f;
  if (i < BN * HIDN) acc[i] = 0.0f;
}

// ---------------------------------------------------------------------------
// Z[bt, :] = emb_table[x[bt]] @ W1[0:300, :] + b1      (bt = b*T + t)
// One wave per 16x16 output tile. Embedding rows staged through LDS.
__global__ void elman_embed_inproj_kernel(const int*   __restrict__ x,
                                          const float* __restrict__ emb,
                                          const float* __restrict__ W1,
                                          const float* __restrict__ b1,
                                          float*       __restrict__ Z) {
  __shared__ float sA[16 * ZSTR];
  __shared__ int   sTok[16];

  const int lane = threadIdx.x;          // 0..31 (wave32)
  const int mt   = blockIdx.x;           // 0..(B*T/16 - 1)
  const int nt   = blockIdx.y;           // 0..18
  const int bt0  = mt * 16;

  if (lane < 16) sTok[lane] = x[bt0 + lane];
  __syncthreads();

  // gather 16 embedding rows (300 f32 each) into LDS
  for (int idx = lane; idx < 16 * EMBN; idx += 32) {
    int r = idx / EMBN;
    int c = idx - r * EMBN;
    sA[r * ZSTR + c] = emb[(long)sTok[r] * EMBN + c];
  }
  __syncthreads();

  const int half = lane >> 4;
  const int mrow = lane & 15;
  const int ncol = nt * 16 + (lane & 15);               // 0..303
  const int nw   = ncol < (HSZ - 1) ? ncol : (HSZ - 1); // clamp pad cols

  v8f c;
  const float bias = b1[nw];
  for (int v = 0; v < 8; ++v) c[v] = bias;

  const float* arow = &sA[mrow * ZSTR];
  for (int kb = 0; kb < HSZ / 4; ++kb) {
    const int kk = kb * 4 + half * 2;
    v2f a = {arow[kk], arow[kk + 1]};
    v2f b = {W1[kk * HSZ + nw], W1[(kk + 1) * HSZ + nw]};
    c = wmma4(a, b, c);
  }
  for (int v = 0; v < 8; ++v) {
    const int row = bt0 + v + half * 8;
    Z[(long)row * ZSTR + ncol] = c[v];
  }
}

// ---------------------------------------------------------------------------
// Pipelined scan step (launch t = 0..T):
//   blocks [0,152):    Hout = sigmoid(Z[:,t] + Hin @ W1[300:600,:])   (t <  T)
//   blocks [152,408):  acc  = max(acc, relu(Hin @ W2 + b2))           (t >  0)
// Both halves read only Hin (written by the previous launch) -> no intra-
// kernel sync needed; stream order is the inter-step barrier.
__global__ void elman_scan_step_kernel(const float* __restrict__ Z,
                                       const float* __restrict__ W1,
                                       const float* __restrict__ W2,
                                       const float* __restrict__ b2,
                                       const float* __restrict__ Hin,
                                       float*       __restrict__ Hout,
                                       float*       __restrict__ acc,
                                       int t) {
  const int lane = threadIdx.x;
  const int half = lane >> 4;
  const int nloc = lane & 15;
  const int id   = blockIdx.x;

  if (id < 8 * NT1) {
    // -------- recurrent matmul + sigmoid: h_{t+1} --------
    if (t >= TN) return;
    const int mt   = id / NT1;
    const int nt   = id - mt * NT1;
    const int ncol = nt * 16 + nloc;                      // 0..303
    const int nw   = ncol < (HSZ - 1) ? ncol : (HSZ - 1);
    const int b0   = mt * 16;

    v8f c;
    for (int v = 0; v < 8; ++v) {
      const int b = b0 + v + half * 8;
      c[v] = Z[((long)b * TN + t) * ZSTR + ncol];         // e_t@W1a + b1
    }
    const float* arow = &Hin[(long)(b0 + nloc) * ZSTR];
    const float* Wr   = &W1[(long)HSZ * HSZ];             // W1[300:600,:]
    for (int kb = 0; kb < HSZ / 4; ++kb) {
      const int kk = kb * 4 + half * 2;
      v2f a  = {arow[kk], arow[kk + 1]};
      v2f bb = {Wr[kk * HSZ + nw], Wr[(kk + 1) * HSZ + nw]};
      c = wmma4(a, bb, c);
    }
    for (int v = 0; v < 8; ++v) {
      const int row = b0 + v + half * 8;
      Hout[(long)row * ZSTR + ncol] = 1.0f / (1.0f + __expf(-c[v]));
    }
  } else {
    // -------- output matmul + relu + running max: out_{t-1} --------
    if (t == 0) return;
    const int id2  = id - 8 * NT1;
    const int mt   = id2 / NT2;
    const int nt   = id2 - mt * NT2;
    const int ncol = nt * 16 + nloc;                      // 0..511
    const int b0   = mt * 16;

    v8f c;
    const float bias = b2[ncol];
    for (int v = 0; v < 8; ++v) c[v] = bias;

    const float* arow = &Hin[(long)(b0 + nloc) * ZSTR];
    for (int kb = 0; kb < HSZ / 4; ++kb) {
      const int kk = kb * 4 + half * 2;
      v2f a  = {arow[kk], arow[kk + 1]};
      v2f bb = {W2[kk * HIDN + ncol], W2[(kk + 1) * HIDN + ncol]};
      c = wmma4(a, bb, c);
    }
    for (int v = 0; v < 8; ++v) {
      const int row = b0 + v + half * 8;
      float r = fmaxf(c[v], 0.0f);
      float* p = &acc[(long)row * HIDN + ncol];
      *p = fmaxf(*p, r);
    }
  }
}

// ---------------------------------------------------------------------------
// out[b, :] = acc[b, :] @ W3 + b3      (128x512 @ 512x4 -> trivial VALU)
__global__ void elman_head_kernel(const float* __restrict__ acc,
                                  const float* __restrict__ W3,
                                  const float* __restrict__ b3,
                                  float* __restrict__ out) {
  const int b = threadIdx.x;  // 0..127
  float s0 = b3[0], s1 = b3[1], s2 = b3[2], s3 = b3[3];
  const float* row = &acc[(long)b * HIDN];
  for (int k = 0; k < HIDN; ++k) {
    const float p = row[k];
    s0 += p * W3[k * NCLSN + 0];
    s1 += p * W3[k * NCLSN + 1];
    s2 += p * W3[k * NCLSN + 2];
    s3 += p * W3[k * NCLSN + 3];
  }
  out[b * NCLSN + 0] = s0;
  out[b * NCLSN + 1] = s1;
  out[b * NCLSN + 2] = s2;
  out[b * NCLSN + 3] = s3;
}

// ---------------------------------------------------------------------------
extern "C" void kernel_launch(void* const* d_in, const int* in_sizes, int n_in,
                              void* d_out, int out_size, void* d_ws, size_t ws_size,
                              hipStream_t stream) {
  (void)in_sizes; (void)n_in; (void)out_size; (void)ws_size;
  const int*   x   = (const int*)  d_in[0];
  const float* emb = (const float*)d_in[1];
  const float* W1  = (const float*)d_in[2];
  const float* b1  = (const float*)d_in[3];
  const float* W2  = (const float*)d_in[4];
  const float* b2  = (const float*)d_in[5];
  const float* W3  = (const float*)d_in[6];
  const float* b3  = (const float*)d_in[7];
  float* out = (float*)d_out;

  // workspace layout (f32): Z [B*T, 304] | h0 [128,304] | h1 [128,304] | acc [128,512]
  float* Z   = (float*)d_ws;
  float* h0  = Z  + (size_t)BN * TN * ZSTR;
  float* h1  = h0 + (size_t)BN * ZSTR;
  float* acc = h1 + (size_t)BN * ZSTR;
  float* hb[2] = {h0, h1};

  elman_init_kernel<<<(BN * HIDN + 255) / 256, 256, 0, stream>>>(h0, acc);

  dim3 gEmb(BN * TN / 16, NT1);
  elman_embed_inproj_kernel<<<gEmb, 32, 0, stream>>>(x, emb, W1, b1, Z);

  for (int t = 0; t <= TN; ++t) {
    elman_scan_step_kernel<<<8 * NT1 + 8 * NT2, 32, 0, stream>>>(
        Z, W1, W2, b2, hb[t & 1], hb[(t + 1) & 1], acc, t);
  }

  elman_head_kernel<<<1, BN, 0, stream>>>(acc, W3, b3, out);
}